// Traj_Conv2d_18442589569598
// MI455X (gfx1250) — compile-verified
//
#include <hip/hip_runtime.h>

#define NN   4
#define CIN  64
#define COUT 128
#define HH   128
#define WW   128
#define HW   (HH*WW)          // 16384
#define KK   9
#define DD   4
#define CD   (CIN*DD)         // 256
#define KP   (CD/2)           // 128 interleaved K-pair rows
#define NOFF 18               // 2*K*K

#define AS1 __attribute__((address_space(1)))
#define AS3 __attribute__((address_space(3)))

typedef __attribute__((ext_vector_type(2))) float v2f;
typedef __attribute__((ext_vector_type(8))) float v8f;
typedef __attribute__((__vector_size__(4*sizeof(int)))) int vi4;

// ---------------------------------------------------------------------------
// Kernel 1: offset field  t = conv3x3(x, w_off) + b_off - x * sum(w_off)
// ---------------------------------------------------------------------------
__global__ __launch_bounds__(256) void offset_conv_kernel(
    const float* __restrict__ x, const float* __restrict__ w_off,
    const float* __restrict__ b_off, float* __restrict__ t)
{
  __shared__ float sW[NOFF*CIN*KK];   // 10368 floats = 41.5 KB
  __shared__ float sWS[NOFF*CIN];     // per (o,c) tap-sum
  __shared__ float sB[NOFF];
  const int tid = threadIdx.x;
  for (int i = tid; i < NOFF*CIN*KK; i += 256) sW[i] = w_off[i];
  if (tid < NOFF) sB[tid] = b_off[tid];
  __syncthreads();
  for (int i = tid; i < NOFF*CIN; i += 256) {
    float s = 0.f;
    #pragma unroll
    for (int q = 0; q < KK; ++q) s += sW[i*KK + q];
    sWS[i] = s;
  }
  __syncthreads();

  const int gid = blockIdx.x*256 + tid;
  const int n = gid / HW;
  const int p = gid % HW;
  const int h = p / WW, w = p % WW;
  const float* xn = x + (size_t)n*CIN*HW;

  float acc[NOFF];
  #pragma unroll
  for (int o = 0; o < NOFF; ++o) acc[o] = sB[o];

  for (int c = 0; c < CIN; ++c) {
    const float* xc = xn + (size_t)c*HW;
    float v[KK];
    #pragma unroll
    for (int ky = 0; ky < 3; ++ky)
      #pragma unroll
      for (int kx = 0; kx < 3; ++kx) {
        const int yy = h + ky - 1, xx = w + kx - 1;
        const bool ok = (yy >= 0) & (yy < HH) & (xx >= 0) & (xx < WW);
        v[ky*3+kx] = ok ? xc[yy*WW + xx] : 0.f;
      }
    const float ctr = v[4];
    #pragma unroll
    for (int o = 0; o < NOFF; ++o) {
      const float* wo = &sW[(o*CIN + c)*KK];
      float s = acc[o];
      #pragma unroll
      for (int q = 0; q < KK; ++q) s = fmaf(wo[q], v[q], s);
      acc[o] = fmaf(-sWS[o*CIN + c], ctr, s);
    }
  }
  float* tn = t + (size_t)n*NOFF*HW + p;
  #pragma unroll
  for (int o = 0; o < NOFF; ++o) tn[(size_t)o*HW] = acc[o];
}

// ---------------------------------------------------------------------------
// Kernel 2: bilinear deformable sampling + depthwise expand (C -> C*D).
// Output y is stored K-PAIR INTERLEAVED for the WMMA GEMM:
//   y[n][kp][p][b] with k = 2*kp + b   (kp in [0,128), b in {0,1})
// so a lane's B operand {k, k+1} is one contiguous 8-byte LDS load.
// ---------------------------------------------------------------------------
__device__ __forceinline__ float gatherv(const float* __restrict__ xc, int yi, int xi) {
  const bool ok = (yi >= 0) & (yi < HH) & (xi >= 0) & (xi < WW);
  const int yy = min(max(yi, 0), HH-1);
  const int xx = min(max(xi, 0), WW-1);
  const float v = xc[yy*WW + xx];
  return ok ? v : 0.f;
}

__global__ __launch_bounds__(256) void deform_dw_kernel(
    const float* __restrict__ x, const float* __restrict__ t,
    const float* __restrict__ w_deform, float* __restrict__ y)
{
  __shared__ float sWd[DD*KK];   // depthwise weights for this channel
  __shared__ float sWs[DD];
  const int bid = blockIdx.x;
  const int hT  = bid % (HH/2);
  const int c   = (bid / (HH/2)) % CIN;
  const int n   = bid / ((HH/2)*CIN);
  const int tid = threadIdx.x;
  if (tid < DD*KK) sWd[tid] = w_deform[(size_t)(c*DD)*KK + tid];
  __syncthreads();
  if (tid < DD) {
    float s = 0.f;
    #pragma unroll
    for (int q = 0; q < KK; ++q) s += sWd[tid*KK + q];
    sWs[tid] = s;
  }
  __syncthreads();

  const int h = hT*2 + (tid / WW);
  const int w = tid % WW;
  const int p = h*WW + w;
  const float* xc = x + ((size_t)n*CIN + c)*HW;
  const float* tn = t + (size_t)n*NOFF*HW + p;

  float acc[DD] = {0.f, 0.f, 0.f, 0.f};
  #pragma unroll
  for (int tap = 0; tap < KK; ++tap) {
    const float dy = tn[(size_t)(2*tap    )*HW];
    const float dx = tn[(size_t)(2*tap + 1)*HW];
    const float py = dy + (float)(h - 1 + tap/3);
    const float px = dx + (float)(w - 1 + tap%3);
    const float fy = floorf(py), fx = floorf(px);
    const float wy = py - fy,    wx = px - fx;
    const int y0 = (int)fy, x0 = (int)fx;
    const float v00 = gatherv(xc, y0,   x0  );
    const float v01 = gatherv(xc, y0,   x0+1);
    const float v10 = gatherv(xc, y0+1, x0  );
    const float v11 = gatherv(xc, y0+1, x0+1);
    const float v = v00*(1.f-wy)*(1.f-wx) + v01*(1.f-wy)*wx
                  + v10*wy*(1.f-wx)       + v11*wy*wx;
    #pragma unroll
    for (int d = 0; d < DD; ++d) acc[d] = fmaf(v, sWd[d*KK + tap], acc[d]);
  }
  const float ctr = xc[p];
  float o0 = fmaf(-sWs[0], ctr, acc[0]);
  float o1 = fmaf(-sWs[1], ctr, acc[1]);
  float o2 = fmaf(-sWs[2], ctr, acc[2]);
  float o3 = fmaf(-sWs[3], ctr, acc[3]);
  // k = c*4+d  ->  kp = c*2 + (d>>1), b = d&1  (two coalesced float2 stores)
  float* yn = y + (size_t)n*CD*HW + (size_t)(c*2)*(HW*2) + (size_t)p*2;
  *(float2*)(yn)               = make_float2(o0, o1);
  *(float2*)(yn + (size_t)HW*2) = make_float2(o2, o3);
}

// ---------------------------------------------------------------------------
// Kernel 3: pointwise GEMM out[n,o,p] = sum_k w_pw[o,k] * y[n,k,p]
//   WMMA f32 16x16x4: A = w_pw tile (M=o), B = y tile (N=pixels).
//   256 threads = 8 waves; wave wv owns o in [wv*16, wv*16+16).
//   Block covers 64 pixels; y tile (128 kp-rows x 128 floats, interleaved)
//   staged in LDS with GLOBAL_LOAD_ASYNC_TO_LDS_B128 (ASYNCcnt).
// ---------------------------------------------------------------------------
__global__ __launch_bounds__(256) void pw_gemm_kernel(
    const float* __restrict__ y, const float* __restrict__ wpw,
    float* __restrict__ out)
{
  __shared__ __align__(16) float ytile[KP*128];   // 65536 B, interleaved layout
  const int tid = threadIdx.x;
  const int blk = blockIdx.x;
  const int n  = blk / (HW/64);
  const int pT = (blk % (HW/64)) * 64;
  // interleaved tile: row kp is 128 contiguous floats starting at pT*2
  const float* yb = y + (size_t)n*CD*HW + (size_t)pT*2;

  if (tid < COUT) __builtin_prefetch(&wpw[(size_t)tid*CD], 0, 0);

  const int lane = tid & 31;
  // cooperative straight copy: one wave moves one 512B row per step
  for (int r = tid >> 5; r < KP; r += 8) {
    const float* src = yb + (size_t)r*(HW*2) + lane*4;
    float*       dst = &ytile[r*128 + lane*4];
#if __has_builtin(__builtin_amdgcn_global_load_async_to_lds_b128)
    __builtin_amdgcn_global_load_async_to_lds_b128(
        (AS1 vi4*)src, (AS3 vi4*)dst, 0, 0);
#else
    *(float4*)dst = *(const float4*)src;
#endif
  }
#if __has_builtin(__builtin_amdgcn_global_load_async_to_lds_b128)
#if __has_builtin(__builtin_amdgcn_s_wait_asynccnt)
  __builtin_amdgcn_s_wait_asynccnt(0);
#else
  asm volatile("s_wait_asynccnt 0x0" ::: "memory");
#endif
#endif
  __syncthreads();

  const int wv   = tid >> 5;        // 0..7
  const int oB   = wv * 16;
  const int l15  = lane & 15;
  const int hi   = lane >> 4;       // 0: K pair {k0,k0+1}, 1: K pair {k0+2,k0+3}

  v8f c0 = {}, c1 = {}, c2 = {}, c3 = {};
  const float* arow = wpw + (size_t)(oB + l15)*CD;   // w_pw row for M=o

  for (int k0 = 0; k0 < CD; k0 += 4) {
    const int ka = k0 + 2*hi;                         // even
    const v2f a = *(const v2f*)(arow + ka);           // A: M=l15, K=ka,ka+1
    // interleaved LDS: {ka, ka+1} at col is one aligned 8-byte pair
    const float* bp = &ytile[(ka >> 1)*128 + l15*2];
    const v2f b0 = *(const v2f*)(bp);
    const v2f b1 = *(const v2f*)(bp + 32);
    const v2f b2 = *(const v2f*)(bp + 64);
    const v2f b3 = *(const v2f*)(bp + 96);
    c0 = __builtin_amdgcn_wmma_f32_16x16x4_f32(false, a, false, b0, (short)0, c0, false, false);
    c1 = __builtin_amdgcn_wmma_f32_16x16x4_f32(false, a, false, b1, (short)0, c1, false, false);
    c2 = __builtin_amdgcn_wmma_f32_16x16x4_f32(false, a, false, b2, (short)0, c2, false, false);
    c3 = __builtin_amdgcn_wmma_f32_16x16x4_f32(false, a, false, b3, (short)0, c3, false, false);
  }

  float* ob = out + (size_t)n*COUT*HW + pT;
  #pragma unroll
  for (int r = 0; r < 8; ++r) {
    const int o = oB + r + 8*hi;                       // C/D row layout
    float* orow = ob + (size_t)o*HW + l15;             // lanes 0-15 coalesced
    orow[0]  = c0[r];
    orow[16] = c1[r];
    orow[32] = c2[r];
    orow[48] = c3[r];
  }
}

// ---------------------------------------------------------------------------
extern "C" void kernel_launch(void* const* d_in, const int* in_sizes, int n_in,
                              void* d_out, int out_size, void* d_ws, size_t ws_size,
                              hipStream_t stream) {
  const float* x        = (const float*)d_in[0];
  const float* w_off    = (const float*)d_in[1];
  const float* b_off    = (const float*)d_in[2];
  const float* w_deform = (const float*)d_in[3];
  const float* w_pw     = (const float*)d_in[4];
  float* out = (float*)d_out;

  float* t_ws = (float*)d_ws;                       // N*18*HW  = 4.7 MB
  float* y_ws = t_ws + (size_t)NN*NOFF*HW;          // N*256*HW = 67 MB (interleaved)

  offset_conv_kernel<<<NN*HW/256, 256, 0, stream>>>(x, w_off, b_off, t_ws);
  deform_dw_kernel<<<NN*CIN*(HH/2), 256, 0, stream>>>(x, t_ws, w_deform, y_ws);
  pw_gemm_kernel<<<NN*HW/64, 256, 0, stream>>>(y_ws, w_pw, out);
}